// MulGONET_45286135169691
// MI455X (gfx1250) — compile-verified
//
#include <hip/hip_runtime.h>
#include <cstdint>
#include <cstddef>

// ---------------------------------------------------------------------------
// Problem constants (from the reference)
// ---------------------------------------------------------------------------
#define INPUT_DIM 20000
#define UNITS     5000
#define NNZ_CNT   1000000
#define BATCH     4096

// LDS strides (floats)
#define A_STRIDE 20    // 16 K-cols padded to 20: 16B-aligned b128 stores, conflict-free b64 reads
#define B_STRIDE 288   // 128 cols * 2 (K-pair interleave) padded +32: half-waves hit disjoint banks

typedef __attribute__((ext_vector_type(2))) float v2f;
typedef __attribute__((ext_vector_type(8))) float v8f;

// ---------------------------------------------------------------------------
// Phase 1: scatter sparse entries into the dense (INPUT_DIM x UNITS) fp32
// kernel in the workspace. Duplicates accumulate (scatter_nd) -> atomicAdd.
// ---------------------------------------------------------------------------
__global__ void scatter_kernel(const float* __restrict__ vals,
                               const int* __restrict__ idx,
                               float* __restrict__ kmat,
                               int nnz) {
  int i = blockIdx.x * blockDim.x + threadIdx.x;
  if (i < nnz) {
    atomicAdd(&kmat[idx[i]], vals[i]);
  }
}

// ---------------------------------------------------------------------------
// Phase 2: out = tanh(X @ K + bias)
//   X: [BATCH, INPUT_DIM] fp32 row-major, K: [INPUT_DIM, UNITS] fp32 row-major
// 128x128 tile per workgroup, 512 threads = 16 wave32, each wave a 32x32
// subtile (2x2 accumulators) via V_WMMA_F32_16X16X4_F32.
//
// LDS: double-buffered. A tile 128x16 (stride 20). B tile stored K-pair
// interleaved: BsP[kp][col] = {B[2kp][col], B[2kp+1][col]} so each B fragment
// is a single aligned ds_load_b64 (no register repacking), matching the A/B
// VGPR layout of WMMA f32 16x16x4 (lanes 0-15: K={0,1}, lanes 16-31: K={2,3}).
// ---------------------------------------------------------------------------
__global__ __launch_bounds__(512) void gemm_bias_tanh_kernel(
    const float* __restrict__ X,
    const float* __restrict__ K,
    const float* __restrict__ bias,
    float* __restrict__ out) {
  __shared__ float As[2][128 * A_STRIDE];  // 2 x 10240 B
  __shared__ float Bs[2][8 * B_STRIDE];    // 2 x  9216 B

  const int tid  = threadIdx.x;
  const int lane = tid & 31;
  const int wave = tid >> 5;            // 0..15
  const int wm   = (wave >> 2) * 32;    // wave M offset in tile
  const int wn   = (wave & 3) * 32;     // wave N offset in tile
  const int half = lane >> 4;           // 0: lanes 0-15, 1: lanes 16-31
  const int lm   = lane & 15;

  const int M0 = blockIdx.y * 128;
  const int N0 = blockIdx.x * 128;

  // A staging: one float4 per thread (128 rows x 16 K)
  const int arow = tid >> 2;            // 0..127
  const int acol = (tid & 3) << 2;      // 0,4,8,12
  // B staging: 2x2 patch per thread (8 row-pairs x 64 col-pairs)
  const int brp  = tid >> 6;            // row pair 0..7
  const int bcb  = (tid & 63) << 1;     // col base 0,2,..,126

  const bool nfull = (N0 + 128) <= UNITS;

  v8f acc[2][2] = {};

  auto loadA = [&](int k, float4& av) {
    av = *(const float4*)(X + (size_t)(M0 + arow) * INPUT_DIM + k + acol);
  };
  auto loadB = [&](int k, float2& q0, float2& q1) {
    const float* p0 = K + (size_t)(k + 2 * brp) * UNITS;
    const float* p1 = p0 + UNITS;
    if (nfull) {
      q0 = *(const float2*)(p0 + N0 + bcb);
      q1 = *(const float2*)(p1 + N0 + bcb);
    } else {
      const int c = N0 + bcb;
      q0.x = (c + 0 < UNITS) ? p0[c + 0] : 0.0f;
      q0.y = (c + 1 < UNITS) ? p0[c + 1] : 0.0f;
      q1.x = (c + 0 < UNITS) ? p1[c + 0] : 0.0f;
      q1.y = (c + 1 < UNITS) ? p1[c + 1] : 0.0f;
    }
  };
  auto stageLDS = [&](int buf, const float4& av, const float2& q0, const float2& q1) {
    *(float4*)&As[buf][arow * A_STRIDE + acol] = av;
    // K-pair interleave: {B[r][c], B[r+1][c], B[r][c+1], B[r+1][c+1]}
    float4 bq;
    bq.x = q0.x; bq.y = q1.x; bq.z = q0.y; bq.w = q1.y;
    *(float4*)&Bs[buf][brp * B_STRIDE + bcb * 2] = bq;
  };

  // --- prologue: stage slab 0 into buffer 0 ---
  {
    float4 av; float2 q0, q1;
    loadA(0, av);
    loadB(0, q0, q1);
    stageLDS(0, av, q0, q1);
  }
  __syncthreads();

  const int KT = INPUT_DIM / 16;        // 1250 slabs, K has no edge

  for (int kt = 0; kt < KT; ++kt) {
    const int buf  = kt & 1;
    const bool more = (kt + 1) < KT;

    float4 av2; float2 q02, q12;
    if (more) {                         // prefetch next slab into registers
      loadA((kt + 1) * 16, av2);
      loadB((kt + 1) * 16, q02, q12);
    }

    // --- compute: 4 K-substeps x (2x2) WMMA tiles on the staged slab ---
#pragma unroll
    for (int ks = 0; ks < 4; ++ks) {
      const int k0 = ks * 4 + 2 * half; // this half-wave's A K-pair base
      const int kp = ks * 2 + half;     // this half-wave's B K-pair row
      v2f afrag[2], bfrag[2];
#pragma unroll
      for (int mi = 0; mi < 2; ++mi) {
        const float* p = &As[buf][(wm + mi * 16 + lm) * A_STRIDE + k0];
        v2f a; a.x = p[0]; a.y = p[1];
        afrag[mi] = a;
      }
#pragma unroll
      for (int ni = 0; ni < 2; ++ni) {
        const int col = wn + ni * 16 + lm;
        bfrag[ni] = *(const v2f*)&Bs[buf][kp * B_STRIDE + col * 2];
      }
#pragma unroll
      for (int mi = 0; mi < 2; ++mi) {
#pragma unroll
        for (int ni = 0; ni < 2; ++ni) {
          acc[mi][ni] = __builtin_amdgcn_wmma_f32_16x16x4_f32(
              /*neg_a=*/false, afrag[mi],
              /*neg_b=*/false, bfrag[ni],
              /*c_mod=*/(short)0, acc[mi][ni],
              /*reuse_a=*/false, /*reuse_b=*/false);
        }
      }
    }

    if (more) {
      stageLDS(buf ^ 1, av2, q02, q12); // write idle buffer: no barrier needed first
    }
    __syncthreads();                    // one barrier per slab
  }

  // --- epilogue: +bias, tanh, store ---
  // C/D layout: VGPR i -> lanes 0-15: M=i, lanes 16-31: M=i+8; N = lane%16.
#pragma unroll
  for (int mi = 0; mi < 2; ++mi) {
#pragma unroll
    for (int ni = 0; ni < 2; ++ni) {
      const int col = N0 + wn + ni * 16 + lm;
      if (col < UNITS) {
        const float bz = bias[col];
#pragma unroll
        for (int i = 0; i < 8; ++i) {
          const int row = M0 + wm + mi * 16 + half * 8 + i;
          out[(size_t)row * UNITS + col] = tanhf(acc[mi][ni][i] + bz);
        }
      }
    }
  }
}

// ---------------------------------------------------------------------------
// Launch.  Inputs (setup_inputs order): x, kernel_vector, bias, nonzero_ind.
// Workspace: dense kernel matrix, INPUT_DIM*UNITS fp32 = 400 MB.
// ---------------------------------------------------------------------------
extern "C" void kernel_launch(void* const* d_in, const int* in_sizes, int n_in,
                              void* d_out, int out_size, void* d_ws, size_t ws_size,
                              hipStream_t stream) {
  const float* x    = (const float*)d_in[0];
  const float* kv   = (const float*)d_in[1];
  const float* bias = (const float*)d_in[2];
  const int*   idx  = (const int*)d_in[3];
  float* out  = (float*)d_out;
  float* kmat = (float*)d_ws;

  // Phase 1: dense kernel = scatter_nd(nonzero_ind, kernel_vector)
  hipMemsetAsync(kmat, 0, (size_t)INPUT_DIM * UNITS * sizeof(float), stream);
  scatter_kernel<<<(NNZ_CNT + 255) / 256, 256, 0, stream>>>(kv, idx, kmat, NNZ_CNT);

  // Phase 2: out = tanh(x @ kmat + bias)
  dim3 grid((UNITS + 127) / 128, BATCH / 128);  // 40 x 32 workgroups
  gemm_bias_tanh_kernel<<<grid, 512, 0, stream>>>(x, kmat, bias, out);
}